// SSDChunkwiseScan_40072044871803
// MI455X (gfx1250) — compile-verified
//
#include <hip/hip_runtime.h>
#include <math.h>

// Problem constants (match reference).
#define B_ 4
#define L_ 4096
#define H_ 16
#define D_ 128
#define C_ 64
#define NC_ (L_ / C_)
#define GATING_Cf 8.0f

typedef float v2f __attribute__((ext_vector_type(2)));
typedef float v4f __attribute__((ext_vector_type(4)));
typedef float v8f __attribute__((ext_vector_type(8)));
typedef int v4i_gcc __attribute__((__vector_size__(4 * sizeof(int))));

// ---------------------------------------------------------------------------
// CDNA5 async global->LDS path (gfx1250 GLOBAL_LOAD_ASYNC_TO_LDS_B128,
// tracked by ASYNCcnt). Prototype confirmed by hipcc diagnostics.
// ---------------------------------------------------------------------------
#if defined(__has_builtin)
#if __has_builtin(__builtin_amdgcn_global_load_async_to_lds_b128)
#define HAVE_ASYNC_LDS 1
#endif
#endif

#ifdef HAVE_ASYNC_LDS
__device__ __forceinline__ void async_copy_b128(const float* gsrc, float* ldst) {
  __builtin_amdgcn_global_load_async_to_lds_b128(
      (__attribute__((address_space(1))) v4i_gcc*)(float*)gsrc,
      (__attribute__((address_space(3))) v4i_gcc*)ldst,
      /*imm offset*/ 0, /*cpol*/ 0);
}
__device__ __forceinline__ void wait_async_le8() {
#if __has_builtin(__builtin_amdgcn_s_wait_asynccnt)
  __builtin_amdgcn_s_wait_asynccnt(8);
#else
  asm volatile("s_wait_asynccnt 0x8" ::: "memory");
#endif
  asm volatile("" ::: "memory");
}
__device__ __forceinline__ void wait_async_0() {
#if __has_builtin(__builtin_amdgcn_s_wait_asynccnt)
  __builtin_amdgcn_s_wait_asynccnt(0);
#else
  asm volatile("s_wait_asynccnt 0x0" ::: "memory");
#endif
  asm volatile("" ::: "memory");
}
#endif

// ---------------------------------------------------------------------------
// Intra-chunk scan. One wave32 per (b, chunk, h). Lane owns 4 of the 128
// state columns. WRITE_OUT=false: emit chunk summaries (total_A, final_h).
// WRITE_OUT=true: recompute + fuse correction, NT-store Y.
// ---------------------------------------------------------------------------
template <bool WRITE_OUT>
__global__ __launch_bounds__(32) void scan_kernel(
    const float* __restrict__ alphap, const float* __restrict__ omegap,
    const float* __restrict__ dtp, const float* __restrict__ Kp,
    const float* __restrict__ Vp, const float* __restrict__ betap,
    const float* __restrict__ rgp, float* __restrict__ totalA,
    float* __restrict__ finalh, float* __restrict__ Yp,
    const float* __restrict__ cstates) {
  const int lane = threadIdx.x;
  const int idx = blockIdx.x;
  const int h = idx % H_;
  const int n = (idx / H_) % NC_;
  const int b = idx / (H_ * NC_);
  const int l0 = n * C_;
  const int dlo = lane * 4;

  const long sbase = ((long)(b * L_ + l0)) * H_ + h;  // scalar index, step 0
  const float* __restrict__ Kbase = Kp + sbase * (long)D_;

  v4f h0 = (v4f)0.f, h1 = (v4f)0.f;
  float cA00 = 1.f, cA01 = 0.f, cA10 = 0.f, cA11 = 1.f;

  v4f S0 = (v4f)0.f, S1 = (v4f)0.f;
  if (WRITE_OUT) {
    const float* cs = cstates + (((long)(b * NC_ + n) * H_ + h) * 2) * D_ + dlo;
    S0 = *(const v4f*)(cs);
    S1 = *(const v4f*)(cs + D_);
  }

#ifdef HAVE_ASYNC_LDS
  __shared__ float kbuf[2][8 * D_];  // double buffer: 8 K-rows per buffer
#pragma unroll
  for (int r = 0; r < 8; ++r)
    async_copy_b128(Kbase + (long)r * (H_ * D_) + dlo, &kbuf[0][r * D_ + dlo]);
#endif

  for (int tb = 0; tb < 8; ++tb) {
#ifdef HAVE_ASYNC_LDS
    if (tb < 7) {
#pragma unroll
      for (int r = 0; r < 8; ++r)
        async_copy_b128(Kbase + (long)((tb + 1) * 8 + r) * (H_ * D_) + dlo,
                        &kbuf[(tb + 1) & 1][r * D_ + dlo]);
      wait_async_le8();  // loads complete in order -> current buffer ready
    } else {
      wait_async_0();
    }
#endif
#pragma unroll
    for (int r = 0; r < 8; ++r) {
      const int s = tb * 8 + r;
      const long si = sbase + (long)s * H_;

      v4f kv;
#ifdef HAVE_ASYNC_LDS
      kv = *(const v4f*)&kbuf[tb & 1][r * D_ + dlo];
#else
      kv = *(const v4f*)(Kbase + (long)s * (H_ * D_) + dlo);
      if (s + 8 < C_)
        __builtin_prefetch(Kbase + (long)(s + 8) * (H_ * D_) + dlo, 0, 0);
#endif
      // wave-uniform per-step scalars (constant-cache s_loads)
      const float al = alphap[si];
      const float om = omegap[si];
      const float dtv = dtp[si];
      const float be = betap[si];
      const float rg = rgp[si];
      const float v0 = Vp[2 * si];
      const float v1 = Vp[2 * si + 1];

      // Cayley discretization + eigenvalue gating (A_bar * scale)
      const float tau = 0.5f * dtv;
      const float ta = tau * al;
      const float to = tau * om;
      const float opa = 1.f + ta, oma = 1.f - ta;
      const float to2 = to * to;
      const float inv_det = 1.f / (opa * opa + to2 + 1e-6f);
      const float eig_sq = (oma * oma + to2) * inv_det;
      const float ex = (GATING_Cf * rg - 1.f) * 0.5f;
      const float scale = __powf(fmaxf(eig_sq, 1e-8f), ex);
      const float a11 = (opa * oma - to2) * inv_det * scale;
      const float a12 = 2.f * to * inv_det * scale;

      // retrieval kTh = h . k : per-lane partials + wave32 xor reduction
      float r0 = h0.x * kv.x + h0.y * kv.y + h0.z * kv.z + h0.w * kv.w;
      float r1 = h1.x * kv.x + h1.y * kv.y + h1.z * kv.z + h1.w * kv.w;
#pragma unroll
      for (int m = 16; m > 0; m >>= 1) {
        r0 += __shfl_xor(r0, m, 32);
        r1 += __shfl_xor(r1, m, 32);
      }
      // selective erasure + rotation + gated write
      const float c0 = be * r0, c1 = be * r1;
      const v4f m0 = h0 - c0 * kv;
      const v4f m1 = h1 - c1 * kv;
      const float g = be * dtv;
      h0 = a11 * m0 + a12 * m1 + (g * v0) * kv;
      h1 = a11 * m1 - a12 * m0 + (g * v1) * kv;

      // cum_A = A_t @ cum_A  (emitted value for this step)
      const float n00 = a11 * cA00 + a12 * cA10;
      const float n01 = a11 * cA01 + a12 * cA11;
      const float n10 = a11 * cA10 - a12 * cA00;
      const float n11 = a11 * cA11 - a12 * cA01;
      cA00 = n00; cA01 = n01; cA10 = n10; cA11 = n11;

      if (WRITE_OUT) {
        // Y = local_h + cum_A @ chunk_state ; streamed -> non-temporal
        const v4f y0 = h0 + cA00 * S0 + cA01 * S1;
        const v4f y1 = h1 + cA10 * S0 + cA11 * S1;
        float* yrow = Yp + (si * 2) * D_ + dlo;
        __builtin_nontemporal_store(y0, (v4f*)yrow);
        __builtin_nontemporal_store(y1, (v4f*)(yrow + D_));
      }
    }
  }

  if (!WRITE_OUT) {
    const long base = (long)(b * NC_ + n) * H_ + h;
    float* fh = finalh + base * 2 * D_ + dlo;
    *(v4f*)fh = h0;
    *(v4f*)(fh + D_) = h1;
    if (lane == 0) {
      float* tA = totalA + base * 4;
      tA[0] = cA00; tA[1] = cA01; tA[2] = cA10; tA[3] = cA11;
    }
  }
}

// ---------------------------------------------------------------------------
// Inter-chunk combine as one 128x128x128 f32 GEMM per (b,h) on WMMA.
//   chunk_states(n) = sum_{m<n} G(n,m) @ final_h(m),  G(n,m)=tA_{n-1}..tA_{m+1}
// P[2n+i][2m+j] = G(n,m)[i][j]  (block lower-triangular, built lane-parallel)
// F[2m+j][d]    = final_h(m)[j][d]
// states        = P @ F  via V_WMMA_F32_16X16X4_F32 (8x8 tiles, 32 K-steps).
// One wave32 per (b,h); LDS = 2 * 64KB.
// ---------------------------------------------------------------------------
__global__ __launch_bounds__(32) void interchunk_wmma_kernel(
    const float* __restrict__ totalA, const float* __restrict__ finalh,
    float* __restrict__ cstates) {
  const int lane = threadIdx.x;
  const int h = blockIdx.x % H_;
  const int b = blockIdx.x / H_;
  const int dlo = lane * 4;

  __shared__ float Pbuf[128 * 128];
  __shared__ float Fbuf[128 * 128];

  // Zero P (upper triangle + row pair n=0 stay zero).
  for (int r = 0; r < 128; ++r)
    *(v4f*)&Pbuf[r * 128 + dlo] = (v4f)0.f;

  // Stage F: row (2m+j) = final_h(m)[j][:].
  for (int r = 0; r < 128; ++r) {
    const float* src =
        finalh + ((long)(b * NC_ + (r >> 1)) * H_ + h) * 2 * D_ + (r & 1) * D_ + dlo;
    *(v4f*)&Fbuf[r * 128 + dlo] = *(const v4f*)src;
  }

  // Lane-parallel prefix-product build of P. Lane owns columns m0=lane,
  // m1=lane+32 (2x2 register blocks, start at zero).
  float c0_00 = 0.f, c0_01 = 0.f, c0_10 = 0.f, c0_11 = 0.f;
  float c1_00 = 0.f, c1_01 = 0.f, c1_10 = 0.f, c1_11 = 0.f;
  const int m0 = lane, m1 = lane + 32;
  for (int n = 1; n < NC_; ++n) {
    const float* tA = totalA + ((long)(b * NC_ + (n - 1)) * H_ + h) * 4;  // uniform
    const float t00 = tA[0], t01 = tA[1], t10 = tA[2], t11 = tA[3];
    // cur = tA @ cur (zeros stay zero)
    float u00 = t00 * c0_00 + t01 * c0_10, u01 = t00 * c0_01 + t01 * c0_11;
    float u10 = t10 * c0_00 + t11 * c0_10, u11 = t10 * c0_01 + t11 * c0_11;
    c0_00 = u00; c0_01 = u01; c0_10 = u10; c0_11 = u11;
    u00 = t00 * c1_00 + t01 * c1_10; u01 = t00 * c1_01 + t01 * c1_11;
    u10 = t10 * c1_00 + t11 * c1_10; u11 = t10 * c1_01 + t11 * c1_11;
    c1_00 = u00; c1_01 = u01; c1_10 = u10; c1_11 = u11;
    // column m = n-1 becomes active with identity
    if (m0 == n - 1) { c0_00 = 1.f; c0_01 = 0.f; c0_10 = 0.f; c0_11 = 1.f; }
    if (m1 == n - 1) { c1_00 = 1.f; c1_01 = 0.f; c1_10 = 0.f; c1_11 = 1.f; }
    // store row pair 2n, 2n+1 for this lane's columns
    *(v2f*)&Pbuf[(2 * n + 0) * 128 + 2 * m0] = (v2f){c0_00, c0_01};
    *(v2f*)&Pbuf[(2 * n + 1) * 128 + 2 * m0] = (v2f){c0_10, c0_11};
    *(v2f*)&Pbuf[(2 * n + 0) * 128 + 2 * m1] = (v2f){c1_00, c1_01};
    *(v2f*)&Pbuf[(2 * n + 1) * 128 + 2 * m1] = (v2f){c1_10, c1_11};
  }

  // GEMM: states = P @ F via v_wmma_f32_16x16x4_f32.
  // A tile layout: lanes 0-15 rows M (K=2k), lanes 16-31 rows M (K=2k+... ):
  //   VGPR0 = K {0 | 2}, VGPR1 = K {1 | 3} split across lane halves.
  const int mrow = lane & 15;
  const int khalf = lane >> 4;  // 0: K={0,1}; 1: K={2,3}
  for (int mt = 0; mt < 8; ++mt) {
    for (int nt = 0; nt < 8; ++nt) {
      v8f acc = (v8f)0.f;
#pragma unroll 4
      for (int kt = 0; kt < 32; ++kt) {
        const int k0 = kt * 4 + 2 * khalf;
        const v2f a = *(const v2f*)&Pbuf[(mt * 16 + mrow) * 128 + k0];
        v2f bv;
        bv.x = Fbuf[(k0 + 0) * 128 + nt * 16 + mrow];
        bv.y = Fbuf[(k0 + 1) * 128 + nt * 16 + mrow];
        acc = __builtin_amdgcn_wmma_f32_16x16x4_f32(
            false, a, false, bv, (short)0, acc, false, false);
      }
      // D layout: element v of lane l -> M = v + 8*(l>>4), N = l&15.
#pragma unroll
      for (int v = 0; v < 8; ++v) {
        const int rrow = mt * 16 + khalf * 8 + v;  // global P row
        const int nn = rrow >> 1, i = rrow & 1;
        float* dst = cstates + ((long)(b * NC_ + nn) * H_ + h) * 2 * D_ +
                     i * D_ + nt * 16 + mrow;
        *dst = acc[v];
      }
    }
  }
}

// ---------------------------------------------------------------------------
extern "C" void kernel_launch(void* const* d_in, const int* in_sizes, int n_in,
                              void* d_out, int out_size, void* d_ws,
                              size_t ws_size, hipStream_t stream) {
  (void)in_sizes; (void)n_in; (void)out_size; (void)ws_size;
  const float* alpha = (const float*)d_in[0];
  const float* omega = (const float*)d_in[1];
  const float* dt    = (const float*)d_in[2];
  const float* K     = (const float*)d_in[3];
  const float* V     = (const float*)d_in[4];
  const float* beta  = (const float*)d_in[5];
  const float* rgate = (const float*)d_in[6];

  float* Y = (float*)d_out;                                   // (B,L,H,2,D)
  float* cstates = Y + (size_t)B_ * L_ * H_ * 2 * D_;         // (B,NC,H,2,D)

  float* totalA = (float*)d_ws;                               // (B,NC,H,4)
  float* finalh = totalA + (size_t)B_ * NC_ * H_ * 4;         // (B,NC,H,2,D)

  const int nscan = B_ * NC_ * H_;  // 4096 waves

  // Pass 1: per-chunk summaries (total_A, final_h)
  scan_kernel<false><<<nscan, 32, 0, stream>>>(
      alpha, omega, dt, K, V, beta, rgate, totalA, finalh, nullptr, nullptr);

  // Pass 2: inter-chunk combine as WMMA GEMM -> chunk_states (second output)
  interchunk_wmma_kernel<<<B_ * H_, 32, 0, stream>>>(totalA, finalh, cstates);

  // Pass 3: recompute intra scan (K re-read hits L2), fuse correction -> Y
  scan_kernel<true><<<nscan, 32, 0, stream>>>(
      alpha, omega, dt, K, V, beta, rgate, nullptr, nullptr, Y, cstates);
}